// V2GridLoss_47966194761960
// MI455X (gfx1250) — compile-verified
//
#include <hip/hip_runtime.h>

typedef __attribute__((ext_vector_type(2))) float v2f;
typedef __attribute__((ext_vector_type(8))) float v8f;

#define NCS 96
#define NB1 123
#define STATE_STRIDE 538   // 4 + 2*123 + 3*96
#define ES 250             // 4 + 2*123
#define DT 0.25f
#define MAXCS 22.17f
#define INV_SBASE (1.0f/1000.0f)
#define ITERS 100
#define TOL 1e-6f

// ---- LDS layout (in floats) ----
#define KT_TILES 31                        // ceil(124/4) K-steps of 4
#define NT_TILES 8                         // 128 bus columns / 16
#define BFRAG_ELEMS (KT_TILES*NT_TILES*64) // 15872 floats per matrix
#define ROW_STRIDE 132                     // 128 data + 4 pad -> conflict-free
#define ARR_ELEMS (16*ROW_STRIDE)          // 2112
#define WAVE_REGION (4*ARR_ELEMS)          // S_re,S_im,v_re,v_im = 8448
#define OFF_BRE 0
#define OFF_BIM BFRAG_ELEMS
#define OFF_WAVE (2*BFRAG_ELEMS)           // 31744
#define OFF_COST (OFF_WAVE + 4*WAVE_REGION) // 65536
#define LDS_FLOATS (OFF_COST + 64)         // 65600 -> 262400 bytes

__global__ __launch_bounds__(128)
void v2g_solver(const float* __restrict__ action, const float* __restrict__ state,
                const float* __restrict__ Kre, const float* __restrict__ Kim,
                const float* __restrict__ Lre, const float* __restrict__ Lim,
                float* __restrict__ out)
{
    extern __shared__ float smem[];
    const int tid  = threadIdx.x;
    const int lane = tid & 31;
    const int wave = tid >> 5;
    const int half = lane >> 4;      // 0 or 1
    const int l15  = lane & 15;
    const int waveRow0 = blockIdx.x * 64 + wave * 16;

    // ---- 1. pack K^T into B-fragment layout: Bfrag[kt][nt][lane][t] = K[n][k]
    for (int i = tid; i < BFRAG_ELEMS; i += 128) {
        int t  = i & 1;
        int ln = (i >> 1) & 31;
        int nt = (i >> 6) & 7;
        int kt = i >> 9;
        int k = kt*4 + t + 2*(ln >> 4);
        int n = nt*16 + (ln & 15);
        float re = 0.f, im = 0.f;
        if (k < NB1 && n < NB1) { re = Kre[n*NB1 + k]; im = Kim[n*NB1 + k]; }
        smem[OFF_BRE + i] = re;
        smem[OFF_BIM + i] = im;
    }

    float* sSre = smem + OFF_WAVE + wave * WAVE_REGION;
    float* sSim = sSre + ARR_ELEMS;
    float* sVre = sSim + ARR_ELEMS;
    float* sVim = sVre + ARR_ELEMS;

    // ---- 2. init S = 0, v = 1 (padded columns included)
    for (int i = lane; i < 16*128; i += 32) {
        int r = i >> 7, k = i & 127;
        int a = r*ROW_STRIDE + k;
        sSre[a] = 0.f; sSim[a] = 0.f; sVre[a] = 1.f; sVim[a] = 0.f;
    }

    // ---- 3. per-row preprocessing: power clamp, cost + user_sat, EV scatter
    for (int r = 0; r < 16; ++r) {
        const int rowg = waveRow0 + r;
        const float* st = state + (size_t)rowg * STATE_STRIDE;
        float price = st[3];
        float acc = 0.f;
        for (int j = lane; j < NCS; j += 32) {
            float a   = action[(size_t)rowg * NCS + j];
            float cap = st[ES + 3*j];
            float tl  = st[ES + 3*j + 1];
            int   bus = (int)st[ES + 3*j + 2];
            float conn = (cap > 0.f) ? 1.f : 0.f;
            float mch  = fminf(22.0f,  conn * (70.f - cap) * (1.f/DT));
            float mdis = fmaxf(-22.0f, conn * (15.f - cap) * (1.f/DT));
            float p = a * MAXCS;                 // both sign branches reduce to this
            p = fmaxf(fminf(p, mch), mdis);
            acc += price * p * DT;               // cost
            float nc  = ceilf((cap + p*DT) * 100.f) * 0.01f;
            float dep = (tl == 1.f) ? 1.f : 0.f;
            float dd  = nc - 70.f;
            acc += -10.f * dep * dd * dd;        // user_sat
            atomicAdd(&sSre[r*ROW_STRIDE + bus], p);   // ds_add_f32 scatter
        }
        for (int m = 1; m < 32; m <<= 1) acc += __shfl_xor(acc, m, 32);
        if (lane == 0) smem[OFF_COST + wave*16 + r] = acc;
    }

    // ---- 4. finalize S = (ap + ev_bus)/S_BASE + i*rp/S_BASE
    for (int i = lane; i < 16*NB1; i += 32) {
        int r = i / NB1, k = i - r*NB1;
        const float* st = state + (size_t)(waveRow0 + r) * STATE_STRIDE;
        int a = r*ROW_STRIDE + k;
        sSre[a] = (st[4 + k] + sSre[a]) * INV_SBASE;
        sSim[a] =  st[4 + NB1 + k]      * INV_SBASE;
    }
    __syncthreads();

    // ---- W preload (padded buses get W=1 so v stays 1 there -> u = 0, no NaN)
    float wre[8], wim[8];
#pragma unroll
    for (int nt = 0; nt < 8; ++nt) {
        int idx = nt*16 + l15;
        wre[nt] = (idx < NB1) ? Lre[idx] : 1.0f;
        wim[nt] = (idx < NB1) ? Lim[idx] : 0.0f;
    }

    v8f accRe[8], accIm[8];

    // ---- 5. fixed-point iterations: vk = K * conj(S/v) + W  via f32 WMMA
    for (int it = 0; it < ITERS; ++it) {
        // seed accumulators with W: D = sum(A*B) + W directly
#pragma unroll
        for (int nt = 0; nt < 8; ++nt)
#pragma unroll
            for (int rr = 0; rr < 8; ++rr) { accRe[nt][rr] = wre[nt]; accIm[nt][rr] = wim[nt]; }

#pragma unroll 1   // keep a single body so accumulators stay pinned (no phi-copy block, <256 VGPRs)
        for (int kt = 0; kt < KT_TILES; ++kt) {
            // A fragment: lane l holds u[row = l&15][k = 4*kt + 2*(l>>4) + t]
            int aoff = l15*ROW_STRIDE + kt*4 + 2*half;
            v2f vre = *(const v2f*)(sVre + aoff);
            v2f vim = *(const v2f*)(sVim + aoff);
            v2f sre = *(const v2f*)(sSre + aoff);
            v2f sim = *(const v2f*)(sSim + aoff);
            v2f a_re, a_im, a_imn;
#pragma unroll
            for (int t = 0; t < 2; ++t) {
                float d  = vre[t]*vre[t] + vim[t]*vim[t];
                float rd = __builtin_amdgcn_rcpf(d);
                float ur = (sre[t]*vre[t] + sim[t]*vim[t]) * rd;  // Re conj(S/v)
                float ui = (sre[t]*vim[t] - sim[t]*vre[t]) * rd;  // Im conj(S/v)
                a_re[t] = ur; a_im[t] = ui; a_imn[t] = -ui;
            }
            const float* bre0 = smem + OFF_BRE + (kt*8)*64 + lane*2;
            const float* bim0 = smem + OFF_BIM + (kt*8)*64 + lane*2;
#pragma unroll
            for (int nt = 0; nt < 8; ++nt) {
                v2f bre = *(const v2f*)(bre0 + nt*64);
                v2f bim = *(const v2f*)(bim0 + nt*64);
                accRe[nt] = __builtin_amdgcn_wmma_f32_16x16x4_f32(false, a_re,  false, bre, (short)0, accRe[nt], false, false);
                accRe[nt] = __builtin_amdgcn_wmma_f32_16x16x4_f32(false, a_imn, false, bim, (short)0, accRe[nt], false, false);
                accIm[nt] = __builtin_amdgcn_wmma_f32_16x16x4_f32(false, a_re,  false, bim, (short)0, accIm[nt], false, false);
                accIm[nt] = __builtin_amdgcn_wmma_f32_16x16x4_f32(false, a_im,  false, bre, (short)0, accIm[nt], false, false);
            }
        }

        // convergence diff vs old |v| (re-read from LDS), write back new v
        float diff = 0.f;
#pragma unroll
        for (int nt = 0; nt < 8; ++nt) {
            int n = nt*16 + l15;
#pragma unroll
            for (int rr = 0; rr < 8; ++rr) {
                int row = rr + 8*half;                 // D layout: VGPR rr, half picks +8
                int a = row*ROW_STRIDE + n;
                float ovr = sVre[a], ovi = sVim[a];
                float om  = __builtin_amdgcn_sqrtf(ovr*ovr + ovi*ovi);
                float vr  = accRe[nt][rr];
                float vi  = accIm[nt][rr];
                float nm  = __builtin_amdgcn_sqrtf(vr*vr + vi*vi);
                diff = fmaxf(diff, fabsf(nm - om));
                sVre[a] = vr;
                sVim[a] = vi;
            }
        }
#pragma unroll
        for (int m = 1; m < 32; m <<= 1) diff = fmaxf(diff, __shfl_xor(diff, m, 32));
        // force scalar branch so EXEC stays all-ones for WMMA
        float dU = __uint_as_float(__builtin_amdgcn_readfirstlane(__float_as_uint(diff)));
        if (dU < TOL) break;
    }

    // ---- 6. final loss: 1000 * sum min(0, 0.05 - |1-|v||) + (costs + user_sat)
#pragma unroll
    for (int rr = 0; rr < 8; ++rr) {
        int row = rr + 8*half;
        float vl = 0.f;
#pragma unroll
        for (int nt = 0; nt < 8; ++nt) {
            int n = nt*16 + l15;
            if (n < NB1) {
                int a = row*ROW_STRIDE + n;
                float vr = sVre[a], vi = sVim[a];
                float m  = __builtin_amdgcn_sqrtf(vr*vr + vi*vi);
                vl += fminf(0.f, 0.05f - fabsf(1.f - m));
            }
        }
#pragma unroll
        for (int m = 1; m < 16; m <<= 1) vl += __shfl_xor(vl, m, 32);  // within 16-lane half
        if (l15 == 0)
            out[waveRow0 + row] = 1000.f * vl + smem[OFF_COST + wave*16 + row];
    }
}

extern "C" void kernel_launch(void* const* d_in, const int* in_sizes, int n_in,
                              void* d_out, int out_size, void* d_ws, size_t ws_size,
                              hipStream_t stream) {
    const float* action = (const float*)d_in[0];
    const float* state  = (const float*)d_in[1];
    const float* Kre    = (const float*)d_in[2];
    const float* Kim    = (const float*)d_in[3];
    const float* Lre    = (const float*)d_in[4];
    const float* Lim    = (const float*)d_in[5];
    float* out = (float*)d_out;

    const int B = in_sizes[0] / NCS;          // 32768
    const size_t ldsBytes = LDS_FLOATS * sizeof(float);  // 262400 B < 320 KB
    hipFuncSetAttribute((const void*)v2g_solver,
                        hipFuncAttributeMaxDynamicSharedMemorySize, (int)ldsBytes);
    dim3 grid(B / 64);    // 4 waves/block * 16 rows/wave
    dim3 block(128);
    v2g_solver<<<grid, block, ldsBytes, stream>>>(action, state, Kre, Kim, Lre, Lim, out);
}